// DQN_40046275067975
// MI455X (gfx1250) — compile-verified
//
#include <hip/hip_runtime.h>

typedef __attribute__((ext_vector_type(2))) float v2f;
typedef __attribute__((ext_vector_type(8))) float v8f;

#define NNODE 256
#define DPAD  16

// One workgroup per batch. 256 threads = 8 waves (wave32).
__launch_bounds__(256)
__global__ void dqn_s2v_fused(const float* __restrict__ features,   // [B,256,5]
                              const float* __restrict__ nl,         // [B,256,256]
                              const float* __restrict__ W1, const float* __restrict__ b1,
                              const float* __restrict__ W2, const float* __restrict__ b2,
                              const float* __restrict__ W3, const float* __restrict__ b3,
                              const float* __restrict__ W4, const float* __restrict__ b4,
                              const float* __restrict__ W5, const float* __restrict__ b5,
                              const float* __restrict__ W6, const float* __restrict__ b6,
                              const float* __restrict__ W7, const float* __restrict__ b7,
                              const int*   __restrict__ iterPtr,
                              float* __restrict__ out)               // [B,256]
{
    __shared__ unsigned connBits[NNODE][8];     //  8 KB: row i, bit j = (nl[i][j] > 0)
    __shared__ float    muS[NNODE][DPAD];       // 16 KB: mu, padded cols [5..15] stay 0
    __shared__ float    cmS[NNODE][DPAD];       // 16 KB: conn @ mu result
    __shared__ float    baseS[NNODE][8];        //  8 KB: term1 + term3 per node
    __shared__ float    gsum[5];

    const int b    = blockIdx.x;
    const int tid  = threadIdx.x;      // node index j for per-node work
    const int lane = tid & 31;
    const int wv   = tid >> 5;         // wave id 0..7 (= bitmask dword owned in stage 1)

    // ---------------- Stage 1: single streaming pass over nodeLocations[b] ----------------
    float w4v[5], b4v[5];
#pragma unroll
    for (int d = 0; d < 5; ++d) { w4v[d] = W4[d]; b4v[d] = b4[d]; }

    float s[5] = {0.f, 0.f, 0.f, 0.f, 0.f};     // column sums for term3 (thread = column j)
    const float* nlb = nl + (size_t)b * NNODE * NNODE;

#pragma unroll 4
    for (int i = 0; i < NNODE; ++i) {
        float x = nlb[(size_t)i * NNODE + tid];             // coalesced: contiguous in j
        unsigned long long m = __ballot(x > 0.0f);          // wave32: low 32 bits valid
        if (lane == 0) connBits[i][wv] = (unsigned)m;
#pragma unroll
        for (int d = 0; d < 5; ++d)
            s[d] += fmaxf(fmaf(x, w4v[d], b4v[d]), 0.0f);
    }

    // term1 + term3 for node j = tid
    {
        float f5[5];
#pragma unroll
        for (int d = 0; d < 5; ++d)
            f5[d] = features[((size_t)b * NNODE + tid) * 5 + d];
#pragma unroll
        for (int o = 0; o < 5; ++o) {
            float t1 = b1[o], t3 = b3[o];
#pragma unroll
            for (int k = 0; k < 5; ++k) {
                t1 = fmaf(W1[o * 5 + k], f5[k], t1);
                t3 = fmaf(W3[o * 5 + k], s[k],  t3);
            }
            baseS[tid][o] = t1 + t3;
        }
    }
    // init mu = 0 (padding columns must stay 0 so WMMA N-padding is exact)
#pragma unroll
    for (int n = 0; n < DPAD; ++n) muS[tid][n] = 0.0f;
    __syncthreads();

    // ---------------- Stage 2: embedding iterations, all in LDS ----------------
    const int iters     = *iterPtr;
    const int rowInTile = lane & 15;
    const int khalf     = lane >> 4;            // 0: K=k0,k0+1   1: K=k0+2,k0+3
    float mu5[5] = {0.f, 0.f, 0.f, 0.f, 0.f};

    for (int it = 0; it < iters; ++it) {
        // conn @ mu : f32 WMMA 16x16x4, exact precision.
        // Each wave computes tiles (wv) and (wv+8): rows tile*16 .. tile*16+15, cols 0..15.
        for (int tt = 0; tt < 2; ++tt) {
            const int tile = wv + tt * 8;
            const int row  = tile * 16 + rowInTile;     // A-matrix row for this lane
            v8f acc = {};
#pragma unroll 1
            for (int kd = 0; kd < 8; ++kd) {            // one bitmask dword = 32 K values
                const unsigned bits = connBits[row][kd];
#pragma unroll
                for (int ks = 0; ks < 8; ++ks) {        // 8 WMMA steps of K=4 per dword
                    const int k0 = kd * 32 + ks * 4;
                    const int kb = k0 + khalf * 2;
                    const unsigned bsel = bits >> (ks * 4 + khalf * 2);
                    v2f a;
                    a.x = (bsel & 1u) ? 1.0f : 0.0f;    // A[row][kb]
                    a.y = (bsel & 2u) ? 1.0f : 0.0f;    // A[row][kb+1]
                    v2f bm;
                    bm.x = muS[kb][rowInTile];          // B[kb  ][n]
                    bm.y = muS[kb + 1][rowInTile];      // B[kb+1][n]
                    acc = __builtin_amdgcn_wmma_f32_16x16x4_f32(
                        false, a, false, bm, (short)0, acc, false, false);
                }
            }
            // C/D layout: lane holds (m = v + 8*khalf, n = lane&15) in VGPR v
#pragma unroll
            for (int v = 0; v < 8; ++v)
                cmS[tile * 16 + v + 8 * khalf][rowInTile] = acc[v];
        }
        __syncthreads();

        // mu = relu(term1 + term3 + W2 @ (conn@mu) + b2)   (node j = tid)
#pragma unroll
        for (int o = 0; o < 5; ++o) {
            float t2 = b2[o];
#pragma unroll
            for (int k = 0; k < 5; ++k)
                t2 = fmaf(W2[o * 5 + k], cmS[tid][k], t2);
            mu5[o] = fmaxf(baseS[tid][o] + t2, 0.0f);
        }
        __syncthreads();        // all WMMA reads of old mu are done
#pragma unroll
        for (int o = 0; o < 5; ++o) muS[tid][o] = mu5[o];
        __syncthreads();
    }

    // ---------------- Stage 3: readout ----------------
    if (tid < 5) gsum[tid] = 0.0f;
    __syncthreads();
#pragma unroll
    for (int o = 0; o < 5; ++o) atomicAdd(&gsum[o], mu5[o]);   // ds_add_f32
    __syncthreads();

    float t6[5], t7[5];
#pragma unroll
    for (int o = 0; o < 5; ++o) {
        float a6 = b6[o], a7 = b7[o];
#pragma unroll
        for (int k = 0; k < 5; ++k) {
            a6 = fmaf(W6[o * 5 + k], gsum[k], a6);
            a7 = fmaf(W7[o * 5 + k], mu5[k],  a7);
        }
        t6[o] = fmaxf(a6, 0.0f);
        t7[o] = fmaxf(a7, 0.0f);
    }
    float r = b5[0];
#pragma unroll
    for (int o = 0; o < 5; ++o)
        r = fmaf(W5[o], t6[o], fmaf(W5[5 + o], t7[o], r));
    out[(size_t)b * NNODE + tid] = r;
}

extern "C" void kernel_launch(void* const* d_in, const int* in_sizes, int n_in,
                              void* d_out, int out_size, void* d_ws, size_t ws_size,
                              hipStream_t stream) {
    const float* features = (const float*)d_in[0];
    const float* nl       = (const float*)d_in[1];
    const float* W1 = (const float*)d_in[2];  const float* b1 = (const float*)d_in[3];
    const float* W2 = (const float*)d_in[4];  const float* b2 = (const float*)d_in[5];
    const float* W3 = (const float*)d_in[6];  const float* b3 = (const float*)d_in[7];
    const float* W4 = (const float*)d_in[8];  const float* b4 = (const float*)d_in[9];
    const float* W5 = (const float*)d_in[10]; const float* b5 = (const float*)d_in[11];
    const float* W6 = (const float*)d_in[12]; const float* b6 = (const float*)d_in[13];
    const float* W7 = (const float*)d_in[14]; const float* b7 = (const float*)d_in[15];
    const int*   it = (const int*)d_in[16];
    float* out = (float*)d_out;

    const int B = in_sizes[0] / (NNODE * 5);   // 512
    dqn_s2v_fused<<<dim3(B), dim3(NNODE), 0, stream>>>(
        features, nl, W1, b1, W2, b2, W3, b3, W4, b4, W5, b5, W6, b6, W7, b7, it, out);
}